// BWNet_52561809768762
// MI455X (gfx1250) — compile-verified
//
#include <hip/hip_runtime.h>
#include <hip/hip_bf16.h>

#define N_ENT 8192
#define EMB   256
#define KDIM  32
#define AR_DIM 1024
#define UTYPE 233
#define MAX_STEPS 64

typedef float v2f __attribute__((ext_vector_type(2)));
typedef float v8f __attribute__((ext_vector_type(8)));
typedef float f4  __attribute__((ext_vector_type(4)));

__device__ __forceinline__ float wsum32(float v) {
#pragma unroll
  for (int s = 16; s > 0; s >>= 1) v += __shfl_xor(v, s, 32);
  return v;
}

__device__ __forceinline__ float sigmoidf(float x) {
  return 1.0f / (1.0f + __expf(-x));
}

// ---------------------------------------------------------------------------
// Kernel Z: zero the whole output (268 MB) with non-temporal stores.
// ---------------------------------------------------------------------------
__global__ void zero_out_kernel(f4* __restrict__ p, long n4) {
  long i = (long)blockIdx.x * blockDim.x + threadIdx.x;
  const long stride = (long)gridDim.x * blockDim.x;
  f4 z = {0.f, 0.f, 0.f, 0.f};
  for (; i < n4; i += stride) __builtin_nontemporal_store(z, p + i);
}

// ---------------------------------------------------------------------------
// Kernel K: keys[32][8192] = W_k(32x256) @ E^T(256x8192) + b_k, exact f32 WMMA.
// One wave per 16x16 tile: 2 M-tiles x 512 N-tiles = 1024 waves.
// blockDim = 64 (2 waves), grid = 512 blocks.
// ---------------------------------------------------------------------------
__global__ void __launch_bounds__(64)
keys_wmma_kernel(const float* __restrict__ Wk, const float* __restrict__ bk,
                 const float* __restrict__ E, float* __restrict__ keys) {
  __shared__ float eLds[2][16 * EMB];  // 32 KB
  const int wlocal = threadIdx.x >> 5;
  const int lane   = threadIdx.x & 31;
  const int wglob  = blockIdx.x * 2 + wlocal;
  const int m0 = (wglob & 1) * 16;
  const int n0 = (wglob >> 1) * 16;

  // Stage E rows [n0, n0+16) into LDS with coalesced float4 loads.
  const f4* Ev = reinterpret_cast<const f4*>(E + (size_t)n0 * EMB);
  f4* Lv = reinterpret_cast<f4*>(eLds[wlocal]);
#pragma unroll 4
  for (int i = lane; i < 16 * (EMB / 4); i += 32) Lv[i] = Ev[i];
  __syncthreads();

  const int row = lane & 15;
  const int kb  = (lane < 16) ? 0 : 2;  // f32 A/B fragment K-base per lane half
  const float* wkrow = Wk + (size_t)(m0 + row) * EMB + kb;
  const float* erow  = eLds[wlocal] + row * EMB + kb;

  v8f acc = {};
#pragma unroll 8
  for (int kk = 0; kk < EMB; kk += 4) {
    v2f a, b;
    a[0] = wkrow[kk]; a[1] = wkrow[kk + 1];
    b[0] = erow[kk];  b[1] = erow[kk + 1];
    acc = __builtin_amdgcn_wmma_f32_16x16x4_f32(
        /*neg_a=*/false, a, /*neg_b=*/false, b,
        /*c_mod=*/(short)0, acc, /*reuse_a=*/false, /*reuse_b=*/false);
  }

  // C/D layout: VGPR r holds M = r + (lane<16 ? 0 : 8), N = lane&15.
  const int mbase = m0 + ((lane < 16) ? 0 : 8);
  const int nn = n0 + row;
#pragma unroll
  for (int r = 0; r < 8; ++r) {
    const int m = mbase + r;
    keys[(size_t)m * N_ENT + nn] = acc[r] + bk[m];
  }
}

// ---------------------------------------------------------------------------
// Kernel S: the sequential 64-step selection scan. One persistent workgroup,
// 1024 threads (32 waves). All recurrent state + mask + reductions in LDS.
// ---------------------------------------------------------------------------
__global__ void __launch_bounds__(1024)
scan_kernel(const float* __restrict__ keys,
            const float* __restrict__ utype_mask,
            const float* __restrict__ entity_mask,
            const float* __restrict__ ar_in,
            const float* __restrict__ W_fe, const float* __restrict__ b_fe,
            const float* __restrict__ W_a0, const float* __restrict__ b_a0,
            const float* __restrict__ W_a1, const float* __restrict__ b_a1,
            const float* __restrict__ W_f,  const float* __restrict__ b_f,
            const float* __restrict__ W_i0, const float* __restrict__ b_i0,
            const float* __restrict__ W_i1, const float* __restrict__ b_i1,
            const float* __restrict__ W_o,  const float* __restrict__ b_o,
            const float* __restrict__ ln_g, const float* __restrict__ ln_b,
            const float* __restrict__ W_a3, const float* __restrict__ b_a3,
            const int* __restrict__ self_unit_ct,
            float* __restrict__ out_logits,
            float* __restrict__ out_sel,
            float* __restrict__ out_ar) {
  __shared__ float s_mask[N_ENT];        // 32 KB
  __shared__ float s_ar[AR_DIM];
  __shared__ float s_fe[EMB];
  __shared__ float s_i0[EMB];
  __shared__ float s_x[2 * KDIM];
  __shared__ float s_g[4 * KDIM];
  __shared__ float s_q[KDIM], s_h[KDIM], s_qn[KDIM], s_hn[KDIM], s_cent[KDIM];
  __shared__ float s_red[1024];
  __shared__ float s_rmax[1024];
  __shared__ int   s_ridx[1024];
  __shared__ float s_ssum;
  __shared__ int   s_pick, s_valid, s_hit, s_nanfree, s_done;

  const int tid  = threadIdx.x;
  const int lane = tid & 31;
  const int wave = tid >> 5;

  // ---- init ----
#pragma unroll
  for (int k = 0; k < N_ENT / 1024; ++k)
    s_mask[tid + k * 1024] = entity_mask[tid + k * 1024];
  s_ar[tid] = ar_in[tid];
  if (tid < EMB) {  // func_embed = relu(W_fe @ utype_mask + b_fe)
    float acc = b_fe[tid];
    const float* wr = W_fe + (size_t)tid * UTYPE;
    for (int u = 0; u < UTYPE; ++u) acc = fmaf(wr[u], utype_mask[u], acc);
    s_fe[tid] = fmaxf(acc, 0.0f);
  }
  if (tid < KDIM) { s_q[tid] = 0.f; s_h[tid] = 0.f; }
  if (tid == 0) s_done = 0;
  __syncthreads();

  const int steps = min(MAX_STEPS, self_unit_ct[0]);

  for (int step = 0; step < steps; ++step) {
    const int active = !s_done;

    // i0 = relu(W_a0 @ ar + b_a0 + func_embed); wave-per-output, coalesced.
#pragma unroll
    for (int pass = 0; pass < EMB / 32; ++pass) {
      const int o = pass * 32 + wave;
      const float* wr = W_a0 + (size_t)o * AR_DIM;
      float acc = 0.f;
      for (int j = lane; j < AR_DIM; j += 32) acc = fmaf(wr[j], s_ar[j], acc);
      acc = wsum32(acc);
      if (lane == 0) s_i0[o] = fmaxf(acc + b_a0[o] + s_fe[o], 0.f);
    }
    __syncthreads();

    // i1 = relu(W_a1 @ i0 + b_a1) -> x[0..31]; x[32..63] = q
    {
      const int o = wave;
      const float* wr = W_a1 + (size_t)o * EMB;
      float acc = 0.f;
      for (int j = lane; j < EMB; j += 32) acc = fmaf(wr[j], s_i0[j], acc);
      acc = wsum32(acc);
      if (lane == 0) s_x[o] = fmaxf(acc + b_a1[o], 0.f);
    }
    if (tid < KDIM) s_x[KDIM + tid] = s_q[tid];
    __syncthreads();

    // 4 LSTM gate matvecs (32x64 each), one thread per gate-output.
    if (tid < 4 * KDIM) {
      const int gate = tid >> 5;
      const int o = tid & 31;
      const float* W = (gate == 0) ? W_f : (gate == 1) ? W_i0
                     : (gate == 2) ? W_i1 : W_o;
      const float* B = (gate == 0) ? b_f : (gate == 1) ? b_i0
                     : (gate == 2) ? b_i1 : b_o;
      float acc = B[o];
      const float* wr = W + (size_t)o * (2 * KDIM);
#pragma unroll
      for (int j = 0; j < 2 * KDIM; ++j) acc = fmaf(wr[j], s_x[j], acc);
      s_g[tid] = acc;
    }
    __syncthreads();

    // LSTM + LayerNorms on wave 0 (32 lanes = KDIM).
    if (wave == 0) {
      const float g = ln_g[lane], bb = ln_b[lane];
      auto LN = [&](float x) {
        float m = wsum32(x) * (1.f / 32.f);
        float d = x - m;
        float var = wsum32(d * d) * (1.f / 32.f);
        return d * rsqrtf(var + 1e-5f) * g + bb;
      };
      float af  = sigmoidf(s_g[lane]);
      float arm = sigmoidf(s_g[32 + lane]) * tanhf(s_g[64 + lane]);
      float ao  = sigmoidf(s_g[96 + lane]);
      float forget   = LN(af);
      float remember = LN(arm);
      float lno      = LN(ao);
      float nh = remember + forget * s_h[lane];
      s_qn[lane] = tanhf(nh) * lno;
      s_hn[lane] = nh;
    }
    __syncthreads();

    // logits over all 8192 entities (8 per thread), coalesced key reads.
    float vloc[N_ENT / 1024];
    float lsum = 0.f;
    float lmax = -1.f;  // v >= 0 always
    int   limax = 0;
#pragma unroll
    for (int k = 0; k < N_ENT / 1024; ++k) {
      const int e = tid + k * 1024;
      float dot = 0.f;
#pragma unroll 8
      for (int kk = 0; kk < KDIM; ++kk)
        dot = fmaf(keys[(size_t)kk * N_ENT + e], s_qn[kk], dot);
      float sig = sigmoidf(dot);
      float v = __expf(__logf(sig) * (1.f / 0.8f));  // sig^(1/TEMP)
      v = (v == v) ? v : 0.f;                        // nan_to_num
      vloc[k] = v;
      lsum += v;
      if (v > lmax) { lmax = v; limax = e; }
    }
    s_red[tid] = lsum; s_rmax[tid] = lmax; s_ridx[tid] = limax;
    __syncthreads();
    for (int s = 512; s > 0; s >>= 1) {
      if (tid < s) {
        s_red[tid] += s_red[tid + s];
        float v2 = s_rmax[tid + s]; int i2 = s_ridx[tid + s];
        float v1 = s_rmax[tid];     int i1 = s_ridx[tid];
        if (v2 > v1 || (v2 == v1 && i2 < i1)) { s_rmax[tid] = v2; s_ridx[tid] = i2; }
      }
      __syncthreads();
    }
    if (tid == 0) {
      s_ssum  = s_red[0];
      s_valid = (s_red[0] != 0.f);
      s_pick  = s_ridx[0];
    }
    __syncthreads();

    // emit softmax row (full row of 8192, NT stores — write-once data).
    {
      const float inv   = s_valid ? (1.f / s_ssum) : 0.f;
      const float scale = (active && s_valid) ? inv : 0.f;
#pragma unroll
      for (int k = 0; k < N_ENT / 1024; ++k) {
        const int e = tid + k * 1024;
        __builtin_nontemporal_store(vloc[k] * scale,
                                    out_logits + (size_t)step * N_ENT + e);
      }
    }

    // selection / mask / sel / done / q,h updates (wave 0).
    if (wave == 0) {
      const int pick = s_pick;
      float selv = keys[(size_t)lane * N_ENT + pick];
      float m = wsum32(selv) * (1.f / 32.f);
      float c = selv - m;
      s_cent[lane] = c;
      unsigned long long nanmask = __ballot(c != c);
      const int nanfree = (nanmask == 0ull);
      if (lane == 0) {
        const int hit = active && s_valid && (s_mask[pick] != 0.f);
        s_hit = hit;
        s_nanfree = nanfree;
        if (hit) {
          s_mask[pick] = 0.f;
          out_sel[pick] = 1.0f;
          if (!nanfree) s_done = 1;
        }
      }
      if (active) { s_q[lane] = s_qn[lane]; s_h[lane] = s_hn[lane]; }
    }
    __syncthreads();

    // ar += relu(W_a3 @ centered + b_a3) on hit & nan_free (1024 outputs).
    if (s_hit && s_nanfree) {
      float acc = b_a3[tid];
      const float* wr = W_a3 + (size_t)tid * KDIM;
#pragma unroll
      for (int kk = 0; kk < KDIM; ++kk) acc = fmaf(wr[kk], s_cent[kk], acc);
      s_ar[tid] += fmaxf(acc, 0.f);
    }
    __syncthreads();
  }

  out_ar[tid] = s_ar[tid];
}

// ---------------------------------------------------------------------------
extern "C" void kernel_launch(void* const* d_in, const int* in_sizes, int n_in,
                              void* d_out, int out_size, void* d_ws, size_t ws_size,
                              hipStream_t stream) {
  (void)in_sizes; (void)n_in; (void)out_size; (void)ws_size;

  const float* utype_mask = (const float*)d_in[0];
  const float* entity_mask = (const float*)d_in[1];
  const float* E     = (const float*)d_in[2];
  const float* ar_in = (const float*)d_in[3];
  const float* W_fe  = (const float*)d_in[4];
  const float* b_fe  = (const float*)d_in[5];
  const float* W_k   = (const float*)d_in[6];
  const float* b_k   = (const float*)d_in[7];
  const float* W_a0  = (const float*)d_in[8];
  const float* b_a0  = (const float*)d_in[9];
  const float* W_a1  = (const float*)d_in[10];
  const float* b_a1  = (const float*)d_in[11];
  const float* W_f   = (const float*)d_in[12];
  const float* b_f   = (const float*)d_in[13];
  const float* W_i0  = (const float*)d_in[14];
  const float* b_i0  = (const float*)d_in[15];
  const float* W_i1  = (const float*)d_in[16];
  const float* b_i1  = (const float*)d_in[17];
  const float* W_o   = (const float*)d_in[18];
  const float* b_o   = (const float*)d_in[19];
  const float* ln_g  = (const float*)d_in[20];
  const float* ln_b  = (const float*)d_in[21];
  const float* W_a3  = (const float*)d_in[22];
  const float* b_a3  = (const float*)d_in[23];
  const int*   ct    = (const int*)d_in[24];

  float* out = (float*)d_out;
  float* out_sel = out + (size_t)N_ENT * N_ENT;
  float* out_ar  = out_sel + N_ENT;
  float* keys_ws = (float*)d_ws;  // 32 x 8192 floats = 1 MB

  // 1) zero the full output (268 MB, NT stores — dominates runtime anyway)
  const long n4 = (long)(((size_t)N_ENT * N_ENT + N_ENT + AR_DIM) / 4);
  zero_out_kernel<<<4096, 256, 0, stream>>>((f4*)out, n4);

  // 2) keys GEMM (exact f32 WMMA); leaves keys hot in L2 for the scan
  keys_wmma_kernel<<<512, 64, 0, stream>>>(W_k, b_k, E, keys_ws);

  // 3) sequential 64-step selection scan, one persistent workgroup
  scan_kernel<<<1, 1024, 0, stream>>>(keys_ws, utype_mask, entity_mask, ar_in,
                                      W_fe, b_fe, W_a0, b_a0, W_a1, b_a1,
                                      W_f, b_f, W_i0, b_i0, W_i1, b_i1,
                                      W_o, b_o, ln_g, ln_b, W_a3, b_a3,
                                      ct, out, out_sel, out_ar);
}